// HashGridEncoding_81003083202711
// MI455X (gfx1250) — compile-verified
//
#include <hip/hip_runtime.h>
#include <hip/hip_bf16.h>

// ---------------------------------------------------------------------------
// HashGridEncoding for gfx1250 (MI455X).
// CDNA5 paths used:
//   * tensor_load_to_lds (TDM, TENSORcnt)  -- stages tables 0..10 into LDS
//   * global_load_async_to_lds_b128 (ASYNCcnt) -- stages table 11 into LDS
//   * s_wait_asynccnt / s_wait_tensorcnt
//   * ds_load_b64 gathers from LDS-staged tables
//   * global_prefetch for coords
// ---------------------------------------------------------------------------

#define N_POINTS   2097152
#define N_LEVELS   16
#define THREADS    256
#define PTS_PER_T  8
#define BLOCKS     (N_POINTS / (THREADS * PTS_PER_T))   // 1024

// Hash constants (u32 wraparound == low 32 bits of the int64 reference math;
// we only ever keep <= 16 low bits, so this is bit-exact vs jnp int64 + mod).
constexpr unsigned HA  = 2654435761u;
constexpr unsigned HB  = 29675113u;
constexpr unsigned HC  = 123456789u;
constexpr unsigned BC_ = HB * HC;          // wraps (defined for unsigned)
constexpr unsigned ABC = HA * HB * HC;     // wraps

// Levels 0..11 staged in LDS: sum_{l=0..11} 2^(l+1) entries = 2^13-2 = 8190 float2
#define LDS_F2 8190

struct Tables { const float2* t[16]; };

typedef __attribute__((ext_vector_type(4))) unsigned v4u;
typedef __attribute__((ext_vector_type(8))) unsigned v8u;

// Issue one TDM copy: n 8-byte elements from gptr -> LDS byte offset lds_off.
// D# built per CDNA5 ISA 8.3/8.4 (group0 + group1, 2-group form = up-to-2D).
__device__ __forceinline__ void tdm_load_1d(const void* gptr, unsigned lds_off,
                                            unsigned n /* 8B elements */)
{
    const unsigned long long ga = (unsigned long long)(uintptr_t)gptr;
    const v4u g0 = {
        1u,                                     // count=1 (valid), user mode
        lds_off,                                // lds_addr [63:32]
        (unsigned)ga,                           // global_addr [95:64]
        ((unsigned)(ga >> 32) & 0x01FFFFFFu)    // global_addr [120:96]
            | (2u << 30)                        // type=2 ("image") [127:126]
    };
    const v8u g1 = {
        3u << 16,                               // data_size=8B; mask=0; no pad
        (n & 0xFFFFu) << 16,                    // tensor_dim0[15:0]  @ bit48
        (n >> 16) | (1u << 16),                 // tensor_dim0[31:16]; tensor_dim1=1
        (n & 0xFFFFu) << 16,                    // tile_dim0 = n      @ bit112
        1u,                                     // tile_dim1=1; tile_dim2=0
        n,                                      // tensor_dim0_stride low32
        0u, 0u                                  // stride hi / dim1_stride = 0
    };
    asm volatile("tensor_load_to_lds %0, %1" :: "s"(g0), "s"(g1) : "memory");
}

// Process levels [L0..L1] that share one resolution RESF.
template <int L0, int L1>
__device__ __forceinline__ void process_group(float resf,
                                              float cx, float cy, float cz,
                                              const float2* smem,
                                              const Tables& tabs,
                                              float2* outf)
{
    const float csx = (cx + 1.0f) * resf;
    const float csy = (cy + 1.0f) * resf;
    const float csz = (cz + 1.0f) * resf;
    const float fx = floorf(csx), fy = floorf(csy), fz = floorf(csz);
    const float tx = csx - fx,    ty = csy - fy,    tz = csz - fz;
    const unsigned ix = (unsigned)(int)fx;
    const unsigned iy = (unsigned)(int)fy;
    const unsigned iz = (unsigned)(int)fz;

    // gC = (((x*A + y)*B + z) * C) mod 2^32 ; corner c adds dx*ABC+dy*BC+dz*C
    const unsigned gc = ((ix * HA + iy) * HB + iz) * HC;
    const unsigned h[8] = { gc,             gc + ABC,
                            gc + BC_,       gc + ABC + BC_,
                            gc + HC,        gc + ABC + HC,
                            gc + BC_ + HC,  gc + ABC + BC_ + HC };

    const float ux = 1.0f - tx, uy = 1.0f - ty, uz = 1.0f - tz;
    const float w[8] = { ux*uy*uz, tx*uy*uz, ux*ty*uz, tx*ty*uz,
                         ux*uy*tz, tx*uy*tz, ux*ty*tz, tx*ty*tz };

#pragma unroll
    for (int l = L0; l <= L1; ++l) {
        const unsigned mask = (2u << l) - 1u;      // table size 2^(l+1)
        float ax = 0.0f, ay = 0.0f;
#pragma unroll
        for (int c = 0; c < 8; ++c) {
            const unsigned idx = h[c] & mask;
            float2 f;
            if (l < 12) {
                f = smem[((2u << l) - 2u) + idx];  // LDS-staged (ds_load_b64)
            } else {
                f = tabs.t[l][idx];                // L2-resident (global_load_b64)
            }
            ax = fmaf(w[c], f.x, ax);
            ay = fmaf(w[c], f.y, ay);
        }
        outf[l] = make_float2(ax, ay);
    }
}

__global__ __launch_bounds__(THREADS)
void hashgrid_encode_kernel(const float* __restrict__ coords,
                            Tables tabs,
                            float* __restrict__ out)
{
    __shared__ float2 smem[LDS_F2];                // 65,520 B

    const unsigned tid = threadIdx.x;
    const unsigned lds_base = (unsigned)(uintptr_t)(&smem[0]);

    // ---- Stage tables 0..10 via Tensor Data Mover (wave 0 issues; EXEC
    //      all-ones in that wave; other waves branch around on EXECZ) ----
    if (tid < 32) {
#pragma unroll
        for (int lv = 0; lv < 11; ++lv)
            tdm_load_1d(tabs.t[lv],
                        lds_base + ((2u << lv) - 2u) * 8u,
                        2u << lv);
    }

    // ---- Stage table 11 (32 KB) via per-lane async copies, all waves ----
    {
        const char* g = (const char*)tabs.t[11];
        const unsigned boff = ((2u << 11) - 2u) * 8u;
#pragma unroll
        for (int j = 0; j < 8; ++j) {              // 2048 chunks of 16B
            const unsigned idx = tid + (unsigned)j * THREADS;
            const unsigned lds_addr = lds_base + boff + idx * 16u;
            const unsigned long long ga =
                (unsigned long long)(uintptr_t)(g + idx * 16u);
            asm volatile("global_load_async_to_lds_b128 %0, %1, off"
                         :: "v"(lds_addr), "v"(ga) : "memory");
        }
    }

    asm volatile("s_wait_asynccnt 0" ::: "memory");
    __builtin_amdgcn_s_wait_tensorcnt(0);
    __syncthreads();

    // ---- Per-point work: 8 points per thread, coalesced within each pass ----
    const int base = (int)blockIdx.x * (THREADS * PTS_PER_T);
#pragma unroll 1
    for (int it = 0; it < PTS_PER_T; ++it) {
        const int p = base + it * THREADS + (int)tid;
        const float* cp = coords + (size_t)p * 3;
        if (it + 1 < PTS_PER_T)
            __builtin_prefetch(cp + 3 * THREADS, 0, 0);    // global_prefetch_b8
        const float cx = cp[0], cy = cp[1], cz = cp[2];

        float2 outf[N_LEVELS];
        process_group<0, 0>( 16.0f, cx, cy, cz, smem, tabs, outf);
        process_group<1, 1>( 32.0f, cx, cy, cz, smem, tabs, outf);
        process_group<2, 2>( 64.0f, cx, cy, cz, smem, tabs, outf);
        process_group<3, 3>(128.0f, cx, cy, cz, smem, tabs, outf);
        process_group<4, 4>(256.0f, cx, cy, cz, smem, tabs, outf);
        process_group<5,15>(512.0f, cx, cy, cz, smem, tabs, outf);  // res saturates

        float4* o = (float4*)(out + (size_t)p * (2 * N_LEVELS));
#pragma unroll
        for (int k = 0; k < 8; ++k)
            o[k] = make_float4(outf[2 * k].x,     outf[2 * k].y,
                               outf[2 * k + 1].x, outf[2 * k + 1].y);
    }
}

extern "C" void kernel_launch(void* const* d_in, const int* in_sizes, int n_in,
                              void* d_out, int out_size, void* d_ws, size_t ws_size,
                              hipStream_t stream)
{
    (void)in_sizes; (void)n_in; (void)out_size; (void)d_ws; (void)ws_size;
    const float* coords = (const float*)d_in[0];
    Tables tabs;
    for (int i = 0; i < 16; ++i)
        tabs.t[i] = (const float2*)d_in[1 + i];

    hashgrid_encode_kernel<<<dim3(BLOCKS), dim3(THREADS), 0, stream>>>(
        coords, tabs, (float*)d_out);
}